// GNN_2_40793599377790
// MI455X (gfx1250) — compile-verified
//
#include <hip/hip_runtime.h>
#include <math.h>

typedef __attribute__((ext_vector_type(2))) float v2f;
typedef __attribute__((ext_vector_type(8))) float v8f;

#define HID 128

// ---------------- degree / normalization ----------------
__global__ __launch_bounds__(256) void k_deg(const int* __restrict__ ei, float* __restrict__ deg, int E) {
    int e = blockIdx.x * 256 + threadIdx.x;
    if (e < E) {
        int dst = ei[(size_t)E + e];
        atomicAdd(&deg[dst], 1.0f);
    }
}

__global__ __launch_bounds__(256) void k_norm(const float* __restrict__ deg, float* __restrict__ dis,
                                              float* __restrict__ selfn, int n) {
    int i = blockIdx.x * 256 + threadIdx.x;
    if (i < n) {
        float d = 1.0f + deg[i];
        dis[i] = rsqrtf(d);
        selfn[i] = 1.0f / d;
    }
}

// ---------------- f32 WMMA GEMM: out[N,128] = A[N,K] @ B[K,128] ----------------
// One wave -> one 16x16 tile. Block = 8 waves covering 16 rows x 128 cols.
// Grid.x = N/16 (N divisible by 16). EXEC is all-ones everywhere (WMMA requirement).
__global__ __launch_bounds__(256) void k_gemm_wmma(const float* __restrict__ A,
                                                   const float* __restrict__ B,
                                                   float* __restrict__ out, int K) {
    int lane = threadIdx.x & 31;
    int wave = threadIdx.x >> 5;
    int row0 = blockIdx.x << 4;
    int col0 = wave << 4;
    int r  = lane & 15;   // row (A) / col (B,D) within tile
    int hi = lane >> 4;   // half-wave select

    const float* Arow = A + (size_t)(row0 + r) * (size_t)K;
    v8f acc = {};
    for (int k = 0; k < K; k += 4) {
        // A 16x4: v0 = K=k (lanes 0-15) / K=k+2 (lanes 16-31); v1 = k+1 / k+3
        v2f a = *(const v2f*)&Arow[k + 2 * hi];
        // B 4x16: row striped across lanes within a VGPR, halves hold K and K+2
        v2f b;
        b.x = B[(size_t)(k + 2 * hi) * HID + col0 + r];
        b.y = B[(size_t)(k + 2 * hi + 1) * HID + col0 + r];
        acc = __builtin_amdgcn_wmma_f32_16x16x4_f32(false, a, false, b, (short)0, acc, false, false);
    }
    // D 16x16 f32: VGPR rr = row rr (lanes 0-15) / row rr+8 (lanes 16-31), col = lane&15
    float* orow = out + (size_t)(row0 + hi * 8) * HID + col0 + r;
#pragma unroll
    for (int rr = 0; rr < 8; ++rr)
        orow[(size_t)rr * HID] = acc[rr];
}

// ---------------- edge scatter: agg[dst] += hw[src] * dis[src]*dis[dst] ----------------
// One wave per edge; lanes cover the 128 features in 4 coalesced strips.
__global__ __launch_bounds__(256) void k_scatter(const float* __restrict__ hw, float* __restrict__ agg,
                                                 const int* __restrict__ ei, const float* __restrict__ dis,
                                                 int E) {
    int wid  = (blockIdx.x * 256 + threadIdx.x) >> 5;  // edge id
    int lane = threadIdx.x & 31;
    if (wid >= E) return;  // wave-uniform
    int s = ei[wid];
    int d = ei[(size_t)E + wid];
    float nrm = dis[s] * dis[d];
    const float* hs = hw + (size_t)s * HID;
    float*       ad = agg + (size_t)d * HID;
#pragma unroll
    for (int i = 0; i < 4; ++i) {
        int f = lane + i * 32;
        atomicAdd(&ad[f], hs[f] * nrm);
    }
}

// ---------------- epilogue: h = relu(agg + hw*selfn + b) ----------------
__global__ __launch_bounds__(256) void k_finalize(const float* __restrict__ agg, const float* __restrict__ hw,
                                                  const float* __restrict__ selfn, const float* __restrict__ b,
                                                  float* __restrict__ out, int total) {
    int idx = blockIdx.x * 256 + threadIdx.x;
    if (idx < total) {
        int n = idx >> 7;
        int f = idx & (HID - 1);
        float v = agg[idx] + hw[idx] * selfn[n] + b[f];
        out[idx] = fmaxf(v, 0.0f);
    }
}

// ---------------- pooling ----------------
__global__ __launch_bounds__(256) void k_counts(const int* __restrict__ batch, float* __restrict__ counts, int n) {
    int i = blockIdx.x * 256 + threadIdx.x;
    if (i < n) atomicAdd(&counts[batch[i]], 1.0f);
}

__global__ __launch_bounds__(256) void k_pool(const float* __restrict__ h, const int* __restrict__ batch,
                                              float* __restrict__ pooled, int total) {
    int idx = blockIdx.x * 256 + threadIdx.x;
    if (idx < total) {
        int n = idx >> 7;
        int f = idx & (HID - 1);
        atomicAdd(&pooled[(size_t)batch[n] * HID + f], h[idx]);
    }
}

// ---------------- BN + linear head; writes xb then out ----------------
__global__ __launch_bounds__(128) void k_head(const float* __restrict__ pooled, const float* __restrict__ counts,
                                              const float* __restrict__ gamma, const float* __restrict__ beta,
                                              const float* __restrict__ mean, const float* __restrict__ var,
                                              const float* __restrict__ lw, const float* __restrict__ lb,
                                              float* __restrict__ dout, int G, int C) {
    __shared__ float xb[HID];
    int g = blockIdx.x;
    int f = threadIdx.x;
    float cnt = fmaxf(counts[g], 1.0f);
    float p   = pooled[(size_t)g * HID + f] / cnt;
    float v   = (p - mean[f]) * rsqrtf(var[f] + 1e-5f) * gamma[f] + beta[f];
    dout[(size_t)g * HID + f] = v;
    xb[f] = v;
    __syncthreads();
    if (f < C) {
        float s = lb[f];
        for (int k = 0; k < HID; ++k) s += xb[k] * lw[k * C + f];
        dout[(size_t)G * HID + (size_t)g * C + f] = fmaxf(s, 0.0f);
    }
}

extern "C" void kernel_launch(void* const* d_in, const int* in_sizes, int n_in,
                              void* d_out, int out_size, void* d_ws, size_t ws_size,
                              hipStream_t stream) {
    const float* x     = (const float*)d_in[0];
    const int*   ei    = (const int*)d_in[1];
    const int*   batch = (const int*)d_in[3];
    const float* W1 = (const float*)d_in[4];
    const float* b1 = (const float*)d_in[5];
    const float* W2 = (const float*)d_in[6];
    const float* b2 = (const float*)d_in[7];
    const float* W3 = (const float*)d_in[8];
    const float* b3 = (const float*)d_in[9];
    const float* gamma = (const float*)d_in[10];
    const float* beta  = (const float*)d_in[11];
    const float* mean  = (const float*)d_in[12];
    const float* var   = (const float*)d_in[13];
    const float* lw    = (const float*)d_in[14];
    const float* lb    = (const float*)d_in[15];

    const int C    = 2;
    const int G    = out_size / (HID + C);     // 128
    const int E    = in_sizes[1] / 2;          // 1,600,000
    const int F_IN = in_sizes[4] / HID;        // 256
    const int N    = in_sizes[0] / F_IN;       // 100,000

    // ---- workspace layout ----
    char* ws = (char*)d_ws;
    size_t off = 0;
    auto wsalloc = [&](size_t bytes) -> void* {
        void* p = ws + off;
        off += (bytes + 255) & ~(size_t)255;
        return p;
    };
    float* deg    = (float*)wsalloc((size_t)N * 4);
    float* dis    = (float*)wsalloc((size_t)N * 4);
    float* selfn  = (float*)wsalloc((size_t)N * 4);
    float* hA     = (float*)wsalloc((size_t)N * HID * 4);  // hw buffer
    float* hB     = (float*)wsalloc((size_t)N * HID * 4);
    float* hC     = (float*)wsalloc((size_t)N * HID * 4);
    float* pooled = (float*)wsalloc((size_t)G * HID * 4);
    float* counts = (float*)wsalloc((size_t)G * 4);
    (void)ws_size;

    const int NE   = N * HID;                 // 12,800,000
    const int gN   = (N + 255) / 256;
    const int gE   = (E + 255) / 256;
    const int gNE  = (NE + 255) / 256;
    const int gROW = N / 16;                  // 6250 (N divisible by 16)
    const int gEDG = (E * 32 + 255) / 256;    // wave per edge

    // ---- degrees & normalization ----
    hipMemsetAsync(deg, 0, (size_t)N * 4, stream);
    k_deg<<<gE, 256, 0, stream>>>(ei, deg, E);
    k_norm<<<gN, 256, 0, stream>>>(deg, dis, selfn, N);

    // ---- layer 1: x[N,256] -> hB ----
    k_gemm_wmma<<<gROW, 256, 0, stream>>>(x, W1, hA, F_IN);
    hipMemsetAsync(hB, 0, (size_t)NE * 4, stream);
    k_scatter<<<gEDG, 256, 0, stream>>>(hA, hB, ei, dis, E);
    k_finalize<<<gNE, 256, 0, stream>>>(hB, hA, selfn, b1, hB, NE);

    // ---- layer 2: hB -> hC ----
    k_gemm_wmma<<<gROW, 256, 0, stream>>>(hB, W2, hA, HID);
    hipMemsetAsync(hC, 0, (size_t)NE * 4, stream);
    k_scatter<<<gEDG, 256, 0, stream>>>(hA, hC, ei, dis, E);
    k_finalize<<<gNE, 256, 0, stream>>>(hC, hA, selfn, b2, hC, NE);

    // ---- layer 3: hC -> hB ----
    k_gemm_wmma<<<gROW, 256, 0, stream>>>(hC, W3, hA, HID);
    hipMemsetAsync(hB, 0, (size_t)NE * 4, stream);
    k_scatter<<<gEDG, 256, 0, stream>>>(hA, hB, ei, dis, E);
    k_finalize<<<gNE, 256, 0, stream>>>(hB, hA, selfn, b3, hB, NE);

    // ---- pooling + head ----
    hipMemsetAsync(pooled, 0, (size_t)G * HID * 4, stream);
    hipMemsetAsync(counts, 0, (size_t)G * 4, stream);
    k_counts<<<gN, 256, 0, stream>>>(batch, counts, N);
    k_pool<<<gNE, 256, 0, stream>>>(hB, batch, pooled, NE);
    k_head<<<G, 128, 0, stream>>>(pooled, counts, gamma, beta, mean, var, lw, lb,
                                  (float*)d_out, G, C);
}